// DeltaBase_34583076667908
// MI455X (gfx1250) — compile-verified
//
#include <hip/hip_runtime.h>
#include <hip/hip_bf16.h>

// Problem constants (match reference)
#define H      64
#define H2     128
#define VOCAB  64
#define BB     256
#define LL     512
#define BL     (BB * LL)

typedef float v2f __attribute__((ext_vector_type(2)));
typedef float v8f __attribute__((ext_vector_type(8)));

// Full-precision WMMA: D = A(16x4 f32) * B(4x16 f32) + C(16x16 f32)
__device__ __forceinline__ v8f wmma4(v2f a, v2f b, v8f c) {
    return __builtin_amdgcn_wmma_f32_16x16x4_f32(
        /*neg_a=*/false, a, /*neg_b=*/false, b,
        /*c_mod=*/(short)0, c, /*reuse_a=*/false, /*reuse_b=*/false);
}

// ---------------------------------------------------------------------------
// Kernel 1: h = LN( embed[seq] + MLP(embed[seq]) )   -> hbuf [B, L, H] fp32
// Block = 128 threads = 4 waves; each wave owns a 16-token tile (64 tok/WG).
// GEMMs via V_WMMA_F32_16X16X4_F32 (K stepped by 4).
// ---------------------------------------------------------------------------
#define HS 68    // h tile LDS stride  (pad: bank-conflict-free A-frag reads)
#define FS 132   // ff tile LDS stride

__global__ __launch_bounds__(128) void k_token(
    const int* __restrict__ seq, const float* __restrict__ embed,
    const float* __restrict__ w1, const float* __restrict__ b1,
    const float* __restrict__ w2, const float* __restrict__ b2,
    const float* __restrict__ ln_g, const float* __restrict__ ln_b,
    float* __restrict__ hbuf)
{
    __shared__ float ht[4][16][HS];   // per-wave 16x64 h tile
    __shared__ float ff[4][16][FS];   // per-wave 16x128 relu intermediate

    const int tid  = threadIdx.x;
    const int wv   = tid >> 5;
    const int lane = tid & 31;
    const int l15  = lane & 15;       // A-row / B,C,D-col within fragment
    const int kh   = lane >> 4;       // half-wave select
    const long tok0 = (long)blockIdx.x * 64 + wv * 16;

    float (*h)[HS] = ht[wv];
    float (*f)[FS] = ff[wv];

    // Gather embeddings for this wave's 16 tokens
    for (int idx = lane; idx < 16 * H; idx += 32) {
        int r = idx >> 6, c = idx & 63;
        int v = seq[tok0 + r];
        h[r][c] = embed[v * H + c];
    }
    __syncthreads();

    // GEMM1: ff = relu(h @ w1 + b1)   [16x64]x[64x128]
    for (int nt = 0; nt < 8; ++nt) {
        const int n0 = nt * 16;
        const float bv = b1[n0 + l15];
        v8f acc;
        #pragma unroll
        for (int i = 0; i < 8; ++i) acc[i] = bv;
        #pragma unroll
        for (int kk = 0; kk < 16; ++kk) {
            const int k0 = kk * 4 + kh * 2;
            v2f a, bf;
            a.x  = h[l15][k0];
            a.y  = h[l15][k0 + 1];
            bf.x = w1[(k0    ) * H2 + n0 + l15];
            bf.y = w1[(k0 + 1) * H2 + n0 + l15];
            acc = wmma4(a, bf, acc);
        }
        #pragma unroll
        for (int i = 0; i < 8; ++i) {
            float v = acc[i];
            f[i + 8 * kh][n0 + l15] = v > 0.f ? v : 0.f;
        }
    }
    __syncthreads();

    // GEMM2 + residual: x = h + (ff @ w2 + b2)   [16x128]x[128x64]
    for (int nt = 0; nt < 4; ++nt) {
        const int n0 = nt * 16;
        const float bv = b2[n0 + l15];
        v8f acc;
        #pragma unroll
        for (int i = 0; i < 8; ++i) acc[i] = bv;
        #pragma unroll
        for (int kk = 0; kk < 32; ++kk) {
            const int k0 = kk * 4 + kh * 2;
            v2f a, bf;
            a.x  = f[l15][k0];
            a.y  = f[l15][k0 + 1];
            bf.x = w2[(k0    ) * H + n0 + l15];
            bf.y = w2[(k0 + 1) * H + n0 + l15];
            acc = wmma4(a, bf, acc);
        }
        #pragma unroll
        for (int i = 0; i < 8; ++i)
            h[i + 8 * kh][n0 + l15] += acc[i];   // residual, in place
    }
    __syncthreads();

    // LayerNorm per row (computed redundantly by both half-waves of a row)
    {
        float mu = 0.f, s2 = 0.f;
        #pragma unroll
        for (int j = 0; j < H; ++j) { float v = h[l15][j]; mu += v; s2 += v * v; }
        mu *= (1.f / H);
        const float var = s2 * (1.f / H) - mu * mu;
        const float rs  = rsqrtf(var + 1e-5f);
        const long t = tok0 + l15;
        for (int j = kh * 32; j < kh * 32 + 32; ++j)
            hbuf[t * H + j] = (h[l15][j] - mu) * rs * ln_g[j] + ln_b[j];
    }
}

// ---------------------------------------------------------------------------
// Kernel 2: delta-rule fast-weight scan. One workgroup (256 thr) per batch
// element; M[64][64] lives in LDS. 511 sequential steps; epilogue computes
// y = M @ h_last.
// ---------------------------------------------------------------------------
__global__ __launch_bounds__(256) void k_scan(
    const float* __restrict__ hbuf, float* __restrict__ ybuf)
{
    __shared__ float M[64][65];
    __shared__ float kv[64];
    __shared__ float part[64][4];
    __shared__ float dv[64];

    const int b   = blockIdx.x;
    const int tid = threadIdx.x;
    const int r   = tid >> 2;   // row 0..63
    const int q   = tid & 3;    // quarter of the K range

    #pragma unroll
    for (int i = 0; i < 16; ++i) M[r][q * 16 + i] = 0.f;
    __syncthreads();

    const float* hb = hbuf + (long)b * LL * H;

    for (int t = 0; t < LL - 1; ++t) {
        if (tid < 64) {
            kv[tid] = hb[t * H + tid];
            __builtin_prefetch(&hb[(t + 1) * H + tid], 0, 3);  // global_prefetch_b8
        }
        __syncthreads();

        // ||k|| computed redundantly per thread (broadcast LDS reads)
        float s2 = 0.f;
        #pragma unroll
        for (int j = 0; j < H; ++j) s2 += kv[j] * kv[j];
        const float rn = 1.f / fmaxf(sqrtf(s2), 1e-12f);

        // vp = M @ kn  (4 threads per row, 16 MACs each)
        float p = 0.f;
        #pragma unroll
        for (int j = 0; j < 16; ++j) p += M[r][q * 16 + j] * kv[q * 16 + j];
        part[r][q] = p;
        __syncthreads();

        if (tid < 64) {
            const float vp = (part[tid][0] + part[tid][1] + part[tid][2] + part[tid][3]) * rn;
            dv[tid] = kv[tid] - vp;   // delta-rule error
        }
        __syncthreads();

        // M += outer(dv, kn)
        const float d = dv[r];
        #pragma unroll
        for (int j = 0; j < 16; ++j) M[r][q * 16 + j] += d * (kv[q * 16 + j] * rn);
        __syncthreads();
    }

    // y = M @ h[:, L-1, :]
    float p = 0.f;
    #pragma unroll
    for (int j = 0; j < 16; ++j) p += M[r][q * 16 + j] * hb[(LL - 1) * H + q * 16 + j];
    part[r][q] = p;
    __syncthreads();
    if (tid < 64)
        ybuf[b * H + tid] = part[tid][0] + part[tid][1] + part[tid][2] + part[tid][3];
}

// ---------------------------------------------------------------------------
// Kernel 3: out = (y @ rp_w + rp_b) @ out_w + out_b     [256x64]->[256x64]
// 16 WGs x 128 thr (4 waves); each WG does 16 rows, each wave a 16-col tile.
// Weights staged in LDS with stride 80 -> half-waves hit disjoint bank sets.
// ---------------------------------------------------------------------------
__global__ __launch_bounds__(128) void k_head(
    const float* __restrict__ ybuf,
    const float* __restrict__ rp_w, const float* __restrict__ rp_b,
    const float* __restrict__ out_w, const float* __restrict__ out_b,
    float* __restrict__ out)
{
    __shared__ float rw[64][80];
    __shared__ float ow[64][80];
    __shared__ float yt[16][68];
    __shared__ float zt[16][68];

    const int tid  = threadIdx.x;
    const int wv   = tid >> 5;
    const int lane = tid & 31;
    const int l15  = lane & 15;
    const int kh   = lane >> 4;
    const int row0 = blockIdx.x * 16;
    const int n0   = wv * 16;

    for (int idx = tid; idx < H * H; idx += 128) {
        int k = idx >> 6, n = idx & 63;
        rw[k][n] = rp_w[idx];
        ow[k][n] = out_w[idx];
    }
    for (int idx = tid; idx < 16 * H; idx += 128) {
        int rr = idx >> 6, c = idx & 63;
        yt[rr][c] = ybuf[(row0 + rr) * H + c];
    }
    __syncthreads();

    // z = y @ rp_w + rp_b
    v8f acc;
    {
        const float bv = rp_b[n0 + l15];
        #pragma unroll
        for (int i = 0; i < 8; ++i) acc[i] = bv;
        #pragma unroll
        for (int kk = 0; kk < 16; ++kk) {
            const int k0 = kk * 4 + kh * 2;
            v2f a, bf;
            a.x  = yt[l15][k0];
            a.y  = yt[l15][k0 + 1];
            bf.x = rw[k0][n0 + l15];
            bf.y = rw[k0 + 1][n0 + l15];
            acc = wmma4(a, bf, acc);
        }
        #pragma unroll
        for (int i = 0; i < 8; ++i) zt[i + 8 * kh][n0 + l15] = acc[i];
    }
    __syncthreads();

    // out = z @ out_w + out_b
    {
        const float bv = out_b[n0 + l15];
        #pragma unroll
        for (int i = 0; i < 8; ++i) acc[i] = bv;
        #pragma unroll
        for (int kk = 0; kk < 16; ++kk) {
            const int k0 = kk * 4 + kh * 2;
            v2f a, bf;
            a.x  = zt[l15][k0];
            a.y  = zt[l15][k0 + 1];
            bf.x = ow[k0][n0 + l15];
            bf.y = ow[k0 + 1][n0 + l15];
            acc = wmma4(a, bf, acc);
        }
        #pragma unroll
        for (int i = 0; i < 8; ++i)
            out[(long)(row0 + i + 8 * kh) * VOCAB + n0 + l15] = acc[i];
    }
}

// ---------------------------------------------------------------------------
extern "C" void kernel_launch(void* const* d_in, const int* in_sizes, int n_in,
                              void* d_out, int out_size, void* d_ws, size_t ws_size,
                              hipStream_t stream) {
    const int*   seq   = (const int*)  d_in[0];
    const float* embed = (const float*)d_in[1];
    const float* w1    = (const float*)d_in[2];
    const float* b1    = (const float*)d_in[3];
    const float* w2    = (const float*)d_in[4];
    const float* b2    = (const float*)d_in[5];
    const float* ln_g  = (const float*)d_in[6];
    const float* ln_b  = (const float*)d_in[7];
    const float* rp_w  = (const float*)d_in[8];
    const float* rp_b  = (const float*)d_in[9];
    const float* out_w = (const float*)d_in[10];
    const float* out_b = (const float*)d_in[11];

    float* hbuf = (float*)d_ws;                  // [B, L, H] fp32 = 32 MiB
    float* ybuf = hbuf + (size_t)BL * H;         // [B, H]    fp32 = 64 KiB

    k_token<<<BL / 64, 128, 0, stream>>>(seq, embed, w1, b1, w2, b2, ln_g, ln_b, hbuf);
    k_scan <<<BB, 256, 0, stream>>>(hbuf, ybuf);
    k_head <<<BB / 16, 128, 0, stream>>>(ybuf, rp_w, rp_b, out_w, out_b, (float*)d_out);
}